// GAT_15281493639207
// MI455X (gfx1250) — compile-verified
//
#include <hip/hip_runtime.h>
#include <math.h>

// ---------------- constants from the reference ----------------
#define IN_C   128
#define HEADS  8
#define HID    8
#define F1     (HEADS * HID)   // 64
#define OUT_C  16
#define NEG_SLOPE 0.2f

typedef float v2f __attribute__((ext_vector_type(2)));
typedef float v8f __attribute__((ext_vector_type(8)));

// ---------------- helpers ----------------
__device__ __forceinline__ unsigned encodeF(float f) {
    unsigned u = __float_as_uint(f);
    return (u & 0x80000000u) ? ~u : (u | 0x80000000u);
}
__device__ __forceinline__ float decodeF(unsigned u) {
    return (u & 0x80000000u) ? __uint_as_float(u & 0x7FFFFFFFu)
                             : __uint_as_float(~u);
}
__device__ __forceinline__ float lrelu(float v) {
    return v > 0.0f ? v : NEG_SLOPE * v;
}

// ---------------- GEMM1: h1[N,64] = x[N,128] @ W1[128,64]  (WMMA f32 16x16x4) ----------------
// One wave32 computes a 16-row x 64-col tile. W1 staged in LDS (32 KB).
__global__ void gat_gemm1_wmma(const float* __restrict__ x,
                               const float* __restrict__ W,
                               float* __restrict__ h1,
                               int N, int nTiles) {
    __shared__ float w[IN_C * F1];
    for (int i = threadIdx.x; i < IN_C * F1; i += blockDim.x) w[i] = W[i];
    __syncthreads();

    const int wavesPerBlk = blockDim.x >> 5;
    const int tile = blockIdx.x * wavesPerBlk + (threadIdx.x >> 5);
    if (tile >= nTiles) return;

    const int lane  = threadIdx.x & 31;
    const int row16 = lane & 15;
    const int kh    = lane >> 4;   // 0: K pair {0,1}, 1: K pair {2,3}

    int rowIn = tile * 16 + row16;
    if (rowIn >= N) rowIn = N - 1;            // clamp (EXEC must stay full for WMMA)
    const float* xr = x + (size_t)rowIn * IN_C;

    v8f acc0 = {}, acc1 = {}, acc2 = {}, acc3 = {};

    for (int k = 0; k < IN_C; k += 4) {
        const int ka = k + 2 * kh;
        v2f a;
        a.x = xr[ka];
        a.y = xr[ka + 1];
        const float* w0 = &w[ka * F1 + row16];
        const float* w1 = &w[(ka + 1) * F1 + row16];
        v2f b0; b0.x = w0[0];  b0.y = w1[0];
        v2f b1; b1.x = w0[16]; b1.y = w1[16];
        v2f b2; b2.x = w0[32]; b2.y = w1[32];
        v2f b3; b3.x = w0[48]; b3.y = w1[48];
        acc0 = __builtin_amdgcn_wmma_f32_16x16x4_f32(false, a, false, b0, (short)0, acc0, false, false);
        acc1 = __builtin_amdgcn_wmma_f32_16x16x4_f32(false, a, false, b1, (short)0, acc1, false, false);
        acc2 = __builtin_amdgcn_wmma_f32_16x16x4_f32(false, a, false, b2, (short)0, acc2, false, false);
        acc3 = __builtin_amdgcn_wmma_f32_16x16x4_f32(false, a, false, b3, (short)0, acc3, false, false);
    }

    // C/D layout: VGPR i -> row (i + 8*kh), col = row16 (+ tile col offset)
    #pragma unroll
    for (int i = 0; i < 8; ++i) {
        const int r = tile * 16 + i + 8 * kh;
        if (r < N) {
            float* o = h1 + (size_t)r * F1 + row16;
            o[0]  = acc0[i];
            o[16] = acc1[i];
            o[32] = acc2[i];
            o[48] = acc3[i];
        }
    }
}

// ---------------- GEMM2: h2[N,16] = o1[N,64] @ W2[64,16]  (WMMA f32 16x16x4) ----------------
__global__ void gat_gemm2_wmma(const float* __restrict__ hin,
                               const float* __restrict__ W,
                               float* __restrict__ h2,
                               int N, int nTiles) {
    __shared__ float w[F1 * OUT_C];
    for (int i = threadIdx.x; i < F1 * OUT_C; i += blockDim.x) w[i] = W[i];
    __syncthreads();

    const int wavesPerBlk = blockDim.x >> 5;
    const int tile = blockIdx.x * wavesPerBlk + (threadIdx.x >> 5);
    if (tile >= nTiles) return;

    const int lane  = threadIdx.x & 31;
    const int row16 = lane & 15;
    const int kh    = lane >> 4;

    int rowIn = tile * 16 + row16;
    if (rowIn >= N) rowIn = N - 1;
    const float* xr = hin + (size_t)rowIn * F1;

    v8f acc = {};
    for (int k = 0; k < F1; k += 4) {
        const int ka = k + 2 * kh;
        v2f a;
        a.x = xr[ka];
        a.y = xr[ka + 1];
        v2f b;
        b.x = w[ka * OUT_C + row16];
        b.y = w[(ka + 1) * OUT_C + row16];
        acc = __builtin_amdgcn_wmma_f32_16x16x4_f32(false, a, false, b, (short)0, acc, false, false);
    }

    #pragma unroll
    for (int i = 0; i < 8; ++i) {
        const int r = tile * 16 + i + 8 * kh;
        if (r < N) h2[(size_t)r * OUT_C + row16] = acc[i];
    }
}

// ---------------- per-node attention dots ----------------
__global__ void att_dot1(const float* __restrict__ h1,
                         const float* __restrict__ attS,
                         const float* __restrict__ attD,
                         float* __restrict__ aS, float* __restrict__ aD, int NH) {
    int i = blockIdx.x * blockDim.x + threadIdx.x;
    if (i >= NH) return;
    const int n = i >> 3, h = i & 7;
    const float* hp = h1 + (size_t)n * F1 + h * HID;
    float s = 0.f, d = 0.f;
    #pragma unroll
    for (int c = 0; c < HID; ++c) {
        const float v = hp[c];
        s += v * attS[h * HID + c];
        d += v * attD[h * HID + c];
    }
    aS[i] = s;
    aD[i] = d;
}

__global__ void att_dot2(const float* __restrict__ h2,
                         const float* __restrict__ attS,
                         const float* __restrict__ attD,
                         float* __restrict__ aS, float* __restrict__ aD, int N) {
    int n = blockIdx.x * blockDim.x + threadIdx.x;
    if (n >= N) return;
    const float* hp = h2 + (size_t)n * OUT_C;
    float s = 0.f, d = 0.f;
    #pragma unroll
    for (int c = 0; c < OUT_C; ++c) {
        const float v = hp[c];
        s += v * attS[c];
        d += v * attD[c];
    }
    aS[n] = s;
    aD[n] = d;
}

// ---------------- layer-1 edge passes (H = 8) ----------------
__global__ void edge_max1(const int* __restrict__ src, const int* __restrict__ dst,
                          const float* __restrict__ aS, const float* __restrict__ aD,
                          unsigned* __restrict__ m, int EH) {
    int i = blockIdx.x * blockDim.x + threadIdx.x;
    if (i >= EH) return;
    const int e = i >> 3, h = i & 7;
    const int s = src[e], d = dst[e];
    const float v = lrelu(aS[s * HEADS + h] + aD[d * HEADS + h]);
    atomicMax(&m[d * HEADS + h], encodeF(v));
}

__global__ void edge_den1(const int* __restrict__ src, const int* __restrict__ dst,
                          const float* __restrict__ aS, const float* __restrict__ aD,
                          const unsigned* __restrict__ m, float* __restrict__ den, int EH) {
    int i = blockIdx.x * blockDim.x + threadIdx.x;
    if (i >= EH) return;
    const int e = i >> 3, h = i & 7;
    const int s = src[e], d = dst[e];
    const float v = lrelu(aS[s * HEADS + h] + aD[d * HEADS + h]);
    const float ex = expf(v - decodeF(m[d * HEADS + h]));
    atomicAdd(&den[d * HEADS + h], ex);
}

__global__ void edge_scat1(const int* __restrict__ src, const int* __restrict__ dst,
                           const float* __restrict__ aS, const float* __restrict__ aD,
                           const unsigned* __restrict__ m, const float* __restrict__ den,
                           const float* __restrict__ h1, float* __restrict__ o1, int EH) {
    int i = blockIdx.x * blockDim.x + threadIdx.x;
    if (i >= EH) return;
    const int e = i >> 3, h = i & 7;
    const int s = src[e], d = dst[e];
    const float v = lrelu(aS[s * HEADS + h] + aD[d * HEADS + h]);
    const float ex = expf(v - decodeF(m[d * HEADS + h]));
    const float alpha = ex / (den[d * HEADS + h] + 1e-16f);
    const float* hp = h1 + (size_t)s * F1 + h * HID;
    float* op = o1 + (size_t)d * F1 + h * HID;
    #pragma unroll
    for (int c = 0; c < HID; ++c) atomicAdd(op + c, alpha * hp[c]);
}

// ---------------- ELU (+ b1) ----------------
__global__ void elu_bias(float* __restrict__ o1, const float* __restrict__ b1, int NF) {
    int i = blockIdx.x * blockDim.x + threadIdx.x;
    if (i >= NF) return;
    const float t = o1[i] + b1[i & (F1 - 1)];
    o1[i] = t > 0.f ? t : (expf(t) - 1.f);
}

// ---------------- layer-2 edge passes (H = 1, C = 16) ----------------
__global__ void edge_max2(const int* __restrict__ src, const int* __restrict__ dst,
                          const float* __restrict__ aS, const float* __restrict__ aD,
                          unsigned* __restrict__ m, int E) {
    int e = blockIdx.x * blockDim.x + threadIdx.x;
    if (e >= E) return;
    const int s = src[e], d = dst[e];
    atomicMax(&m[d], encodeF(lrelu(aS[s] + aD[d])));
}

__global__ void edge_den2(const int* __restrict__ src, const int* __restrict__ dst,
                          const float* __restrict__ aS, const float* __restrict__ aD,
                          const unsigned* __restrict__ m, float* __restrict__ den, int E) {
    int e = blockIdx.x * blockDim.x + threadIdx.x;
    if (e >= E) return;
    const int s = src[e], d = dst[e];
    const float v = lrelu(aS[s] + aD[d]);
    atomicAdd(&den[d], expf(v - decodeF(m[d])));
}

__global__ void edge_scat2(const int* __restrict__ src, const int* __restrict__ dst,
                           const float* __restrict__ aS, const float* __restrict__ aD,
                           const unsigned* __restrict__ m, const float* __restrict__ den,
                           const float* __restrict__ h2, float* __restrict__ out, int EC) {
    int i = blockIdx.x * blockDim.x + threadIdx.x;
    if (i >= EC) return;
    const int e = i >> 4, c = i & 15;
    const int s = src[e], d = dst[e];
    const float v = lrelu(aS[s] + aD[d]);
    const float ex = expf(v - decodeF(m[d]));
    const float alpha = ex / (den[d] + 1e-16f);
    atomicAdd(&out[(size_t)d * OUT_C + c], alpha * h2[(size_t)s * OUT_C + c]);
}

// ---------------- final bias ----------------
__global__ void add_b2(float* __restrict__ out, const float* __restrict__ b2, int NC) {
    int i = blockIdx.x * blockDim.x + threadIdx.x;
    if (i >= NC) return;
    out[i] += b2[i & (OUT_C - 1)];
}

// ---------------- host launcher ----------------
extern "C" void kernel_launch(void* const* d_in, const int* in_sizes, int n_in,
                              void* d_out, int out_size, void* d_ws, size_t ws_size,
                              hipStream_t stream) {
    const int N = in_sizes[0] / IN_C;
    const int E = in_sizes[1] / 2;

    const float* x     = (const float*)d_in[0];
    const int*   ei    = (const int*)  d_in[1];
    const int*   src   = ei;
    const int*   dst   = ei + E;
    const float* W1    = (const float*)d_in[2];
    const float* b1    = (const float*)d_in[3];
    const float* attS1 = (const float*)d_in[4];
    const float* attD1 = (const float*)d_in[5];
    const float* W2    = (const float*)d_in[6];
    const float* b2    = (const float*)d_in[7];
    const float* attS2 = (const float*)d_in[8];
    const float* attD2 = (const float*)d_in[9];
    float* out = (float*)d_out;

    // workspace layout
    char* ws = (char*)d_ws;
    size_t off = 0;
    auto alloc = [&](size_t elems) -> float* {
        float* p = (float*)(ws + off);
        off = (off + elems * sizeof(float) + 255) & ~(size_t)255;
        return p;
    };
    float*    h1   = alloc((size_t)N * F1);
    float*    aS1  = alloc((size_t)N * HEADS);
    float*    aD1  = alloc((size_t)N * HEADS);
    float*    h2   = alloc((size_t)N * OUT_C);
    float*    aS2  = alloc((size_t)N);
    float*    aD2  = alloc((size_t)N);
    const size_t zoff = off;                     // everything below is zero-initialized
    unsigned* m1   = (unsigned*)alloc((size_t)N * HEADS);
    float*    den1 = alloc((size_t)N * HEADS);
    float*    o1   = alloc((size_t)N * F1);
    unsigned* m2   = (unsigned*)alloc((size_t)N);
    float*    den2 = alloc((size_t)N);
    const size_t zbytes = off - zoff;

    hipMemsetAsync(ws + zoff, 0, zbytes, stream);
    hipMemsetAsync(d_out, 0, (size_t)out_size * sizeof(float), stream);

    const int TB = 256;
    auto nb = [](long long n, int tb) { return (unsigned)((n + tb - 1) / tb); };

    const int nTiles = (N + 15) / 16;
    const int gemmBlks = (nTiles + 7) / 8;       // 8 waves/block

    // ---- layer 1 ----
    gat_gemm1_wmma<<<gemmBlks, TB, 0, stream>>>(x, W1, h1, N, nTiles);
    att_dot1<<<nb((long long)N * HEADS, TB), TB, 0, stream>>>(h1, attS1, attD1, aS1, aD1, N * HEADS);
    edge_max1<<<nb((long long)E * HEADS, TB), TB, 0, stream>>>(src, dst, aS1, aD1, m1, E * HEADS);
    edge_den1<<<nb((long long)E * HEADS, TB), TB, 0, stream>>>(src, dst, aS1, aD1, m1, den1, E * HEADS);
    edge_scat1<<<nb((long long)E * HEADS, TB), TB, 0, stream>>>(src, dst, aS1, aD1, m1, den1, h1, o1, E * HEADS);
    elu_bias<<<nb((long long)N * F1, TB), TB, 0, stream>>>(o1, b1, N * F1);

    // ---- layer 2 ----
    gat_gemm2_wmma<<<gemmBlks, TB, 0, stream>>>(o1, W2, h2, N, nTiles);
    att_dot2<<<nb(N, TB), TB, 0, stream>>>(h2, attS2, attD2, aS2, aD2, N);
    edge_max2<<<nb(E, TB), TB, 0, stream>>>(src, dst, aS2, aD2, m2, E);
    edge_den2<<<nb(E, TB), TB, 0, stream>>>(src, dst, aS2, aD2, m2, den2, E);
    edge_scat2<<<nb((long long)E * OUT_C, TB), TB, 0, stream>>>(src, dst, aS2, aD2, m2, den2, h2, out, E * OUT_C);
    add_b2<<<nb((long long)N * OUT_C, TB), TB, 0, stream>>>(out, b2, N * OUT_C);
}